// LocalGNNHGLap_16217796509773
// MI455X (gfx1250) — compile-verified
//
#include <hip/hip_runtime.h>
#include <stdint.h>

// Problem shape (fixed by the reference)
#define BATCH   2
#define FEAT    16
#define CH      32            // BATCH*FEAT == wave32 lanes
#define N_IN    2048
#define N_OUT   4096

// Tiling
#define TN          512                   // n-tile staged in LDS
#define NT          (N_IN / TN)           // 4 tiles
#define WAVES       8
#define MW          2                     // m rows per wave
#define M_PER_BLOCK (WAVES * MW)          // 16
#define XS_STRIDE   514                   // 512 dwords + 2 pad dwords (TDM pad: 1 per 256)
#define XS_DWORDS   (CH * XS_STRIDE)      // 16448 dwords per buffer

typedef float f32x4 __attribute__((ext_vector_type(4)));

// LDS index matching the TDM padded layout: one pad dword per 256 stored dwords.
__device__ __forceinline__ int xidx(int ch, int nn) {
    return ch * XS_STRIDE + nn + (nn >> 8);
}

#if defined(__has_builtin)
#if __has_builtin(__builtin_amdgcn_tensor_load_to_lds) && __has_builtin(__builtin_amdgcn_s_wait_tensorcnt)
#define USE_TDM 1
#endif
#endif

#ifdef USE_TDM
typedef unsigned int u32x4 __attribute__((ext_vector_type(4)));
typedef int          i32x4 __attribute__((ext_vector_type(4)));
typedef int          i32x8 __attribute__((ext_vector_type(8)));

// Issue a TDM load of x[0:32, n0:n0+512] (f32, row stride N_IN) into LDS at dstLds,
// inserting 1 pad dword every 256 dwords (so LDS row pitch = 514 dwords, bank-conflict-free
// for the lane==channel broadcast reads).
__device__ __forceinline__ void tdm_load_x_tile(float* dstLds, const float* xg, int n0) {
    unsigned long long ga  = (unsigned long long)(uintptr_t)(xg + n0);
    unsigned int       lds = (unsigned int)(uintptr_t)dstLds;   // low 32 bits = LDS byte offset

    // D# group 0: count=1 | lds_addr | global_addr[56:0] | type=2 ("image")
    u32x4 g0 = { 1u,
                 lds,
                 (unsigned int)ga,
                 (unsigned int)(ga >> 32) | (2u << 30) };

    // D# group 1:
    //  d0: data_size=2 (4B) | pad_enable | pad_interval=7 (256 dwords) | pad_amount=0 (1 dword)
    //  d1: tensor_dim0 low16 (=2048) in bits[63:48]
    //  d2: tensor_dim1 low16 (=32)   at d2[31:16]
    //  d3: tile_dim0 (=512) in bits[127:112]
    //  d4: tile_dim1 (=32), tile_dim2=0
    //  d5: tensor_dim0_stride (=2048), d6/d7: dim1 stride unused (2D tile)
    i32x8 g1 = { (int)((2u << 16) | (1u << 20) | (7u << 22)),
                 (int)((unsigned)N_IN << 16),
                 (int)((unsigned)CH << 16),
                 (int)((unsigned)TN << 16),
                 (int)CH,
                 (int)N_IN,
                 0, 0 };

    i32x4 g2 = { 0, 0, 0, 0 };
    i32x4 g3 = { 0, 0, 0, 0 };
    i32x8 g4 = { 0, 0, 0, 0, 0, 0, 0, 0 };   // extra group on clang-23 toolchain (zero-filled)

    __builtin_amdgcn_tensor_load_to_lds(g0, g1, g2, g3, g4, 0);
}
#endif

__global__ __launch_bounds__(256)
void pool_clique_max_kernel(const float* __restrict__ x,
                            const float* __restrict__ Bt,
                            float* __restrict__ out) {
    extern __shared__ float xs[];                 // 2 * XS_DWORDS floats (double buffer)
    const int tid   = threadIdx.x;
    const int lane  = tid & 31;                   // channel id AND Bt sub-column id
    const int wave  = tid >> 5;
    const int mBase = blockIdx.x * M_PER_BLOCK + wave * MW;

    float acc[MW];
#pragma unroll
    for (int i = 0; i < MW; ++i) acc[i] = 0.0f;   // dense max includes the x*0 terms

#ifdef USE_TDM
    if (tid < 32) tdm_load_x_tile(xs, x, 0);      // prologue: tile 0 in flight
#endif

    for (int t = 0; t < NT; ++t) {
        float* cur = xs + (t & 1) * XS_DWORDS;

#ifdef USE_TDM
        if (tid < 32) {
            if (t + 1 < NT) {
                tdm_load_x_tile(xs + ((t + 1) & 1) * XS_DWORDS, x, (t + 1) * TN);
                __builtin_amdgcn_s_wait_tensorcnt(1);   // tile t done (TDM ops retire in order)
            } else {
                __builtin_amdgcn_s_wait_tensorcnt(0);
            }
        }
        __syncthreads();
#else
        // Fallback: cooperative row copy into the same padded layout.
        for (int i = tid; i < CH * TN; i += 256) {
            int ch = i >> 9;                      // TN == 512
            int nn = i & (TN - 1);
            cur[xidx(ch, nn)] = x[ch * N_IN + t * TN + nn];
        }
        __syncthreads();
#endif

        const int n0 = t * TN;
#pragma unroll
        for (int mi = 0; mi < MW; ++mi) {
            const int m = mBase + mi;
            const f32x4* bt4 = (const f32x4*)(Bt + (size_t)m * N_IN + n0);
#pragma unroll
            for (int k = 0; k < TN / 128; ++k) {  // wave covers 128 dwords (512 B) per iter
                // Bt is a single-pass 32 MB stream: non-temporal so it never displaces
                // the L2-resident x working set.
                f32x4 v = __builtin_nontemporal_load(&bt4[k * 32 + lane]);
                bool nz = (v.x != 0.0f) | (v.y != 0.0f) | (v.z != 0.0f) | (v.w != 0.0f);
                unsigned long long mask = __ballot(nz);   // wave32: low 32 bits
                while (mask) {
                    int p = __ffsll(mask) - 1;
                    mask &= mask - 1;
                    float vx = __shfl(v.x, p);
                    float vy = __shfl(v.y, p);
                    float vz = __shfl(v.z, p);
                    float vw = __shfl(v.w, p);
                    int nn = k * 128 + p * 4;
                    if (vx != 0.0f) acc[mi] = fmaxf(acc[mi], vx * cur[xidx(lane, nn + 0)]);
                    if (vy != 0.0f) acc[mi] = fmaxf(acc[mi], vy * cur[xidx(lane, nn + 1)]);
                    if (vz != 0.0f) acc[mi] = fmaxf(acc[mi], vz * cur[xidx(lane, nn + 2)]);
                    if (vw != 0.0f) acc[mi] = fmaxf(acc[mi], vw * cur[xidx(lane, nn + 3)]);
                }
            }
        }
        __syncthreads();   // all waves done reading `cur` before it is refilled
    }

    // out[b][c][m] flat = ch * N_OUT + m, lane == ch; write-once -> non-temporal
#pragma unroll
    for (int mi = 0; mi < MW; ++mi) {
        __builtin_nontemporal_store(acc[mi], &out[(size_t)lane * N_OUT + (mBase + mi)]);
    }
}

extern "C" void kernel_launch(void* const* d_in, const int* in_sizes, int n_in,
                              void* d_out, int out_size, void* d_ws, size_t ws_size,
                              hipStream_t stream) {
    (void)in_sizes; (void)n_in; (void)out_size; (void)d_ws; (void)ws_size;
    const float* x  = (const float*)d_in[0];   // [2,16,2048] f32
    const float* Bt = (const float*)d_in[1];   // [4096,2048] f32 binary
    float* out      = (float*)d_out;           // [2,16,4096] f32

    dim3 grid(N_OUT / M_PER_BLOCK);            // 256 blocks
    dim3 block(256);                           // 8 waves
    size_t shmem = 2 * XS_DWORDS * sizeof(float);  // 131,584 B (fits 320 KB/WGP)
    hipLaunchKernelGGL(pool_clique_max_kernel, grid, block, shmem, stream, x, Bt, out);
}